// Inducer_67800353735083
// MI455X (gfx1250) — compile-verified
//
#include <hip/hip_runtime.h>
#include <math.h>

// ---------------------------------------------------------------------------
// CKY beam-search DP ("Inducer"), N=48, BEAM=8, DVEC=62, D2=64.
// Latency-bound sequential DP (47 dependent steps). Strategy:
//  - cache P_fwd = V@Cpad, P_rev = V@CpadT per table entry (WMMA f32 16x16x4)
//  - per split: one 16x16x(K=64) WMMA chain computes both 8x8 bilinear-form
//    matrices F (rows 0-7) and G (rows 8-15) at once
//  - padded tile halves are fed DUPLICATED data (never zero-masked) so the
//    fragment builders are branch-free: discarded outputs need no masking
//  - 4 wave32 per block, split points strided across waves, exact top-8
//    (value desc, tie -> lower index) matching jax.lax.top_k
// ---------------------------------------------------------------------------

typedef __attribute__((ext_vector_type(2))) float v2f;
typedef __attribute__((ext_vector_type(8))) float v8f;

#define NTOK 48
#define DVEC 62
#define TINYF 1e-9f

// workspace layout (in floats). total ~3.59M floats (~13.7 MB)
#define OFF_CPAD  0                       // 64x64 zero-padded cooccurrence
#define OFF_CPADT 4096                    // transpose, zero-padded
#define OFF_ORD6  8192                    // [l0,l1,l2,1-l0,1-l1,1-l2]
#define OFF_VV    8256                    // V[k][i][beam][64] vectors
#define SZ_VV     (48*48*8*64)
#define OFF_PF    (OFF_VV + SZ_VV)        // P_fwd = V @ Cpad
#define OFF_PR    (OFF_PF + SZ_VV)        // P_rev = V @ CpadT
#define OFF_OP    (OFF_PR + SZ_VV)        // op scores  [48*48][8]
#define OFF_ORD   (OFF_OP + 48*48*8)      // ord scores [48*48][8]

__device__ __forceinline__ void wave_lds_fence() {
    // intra-wave LDS producer->consumer ordering (single wave32, lockstep)
    asm volatile("s_wait_dscnt 0x0" ::: "memory");
}

__device__ __forceinline__ bool better(float s, int i, float s2, int i2) {
    return (s > s2) || (s == s2 && i < i2);   // top_k tie-break: lower index
}

struct Top8 { float s[8]; int i[8]; float c[8]; float r[8]; };

__device__ __forceinline__ void top8_reset(Top8& t) {
#pragma unroll
    for (int q = 0; q < 8; ++q) { t.s[q] = -3.4e38f; t.i[q] = 0x7fffffff; t.c[q] = 0.f; t.r[q] = 0.f; }
}

__device__ __forceinline__ void top8_ins(Top8& t, float s, int idx, float cop, float cord) {
    if (better(s, idx, t.s[7], t.i[7])) { t.s[7] = s; t.i[7] = idx; t.c[7] = cop; t.r[7] = cord; }
#pragma unroll
    for (int p = 7; p > 0; --p) {
        if (better(t.s[p], t.i[p], t.s[p-1], t.i[p-1])) {
            float a = t.s[p]; t.s[p] = t.s[p-1]; t.s[p-1] = a;
            int   b = t.i[p]; t.i[p] = t.i[p-1]; t.i[p-1] = b;
            a = t.c[p]; t.c[p] = t.c[p-1]; t.c[p-1] = a;
            a = t.r[p]; t.r[p] = t.r[p-1]; t.r[p-1] = a;
        }
    }
}

// Preload A fragments for the P-product chains: 8 beam rows duplicated into
// the 16-row tile (rows 8-15 produce discarded output), branch-free.
__device__ __forceinline__ void p_preload(const float* vec, int lane, float* a_pre) {
    int row = lane & 7;                 // duplicate rows 8-15 <- rows 0-7
    int hi = lane >> 4;
#pragma unroll
    for (int kk = 0; kk < 16; ++kk) {
        int k0 = kk * 4 + hi * 2;
        a_pre[2 * kk]     = vec[row * 64 + k0];
        a_pre[2 * kk + 1] = vec[row * 64 + k0 + 1];
    }
}

// (8x64 dup-padded to 16) @ (64x64) N-tile nt -> out rows 0..7, cols nt*16..+15
// one wave32; 16 x V_WMMA_F32_16X16X4_F32 accumulation chain
__device__ __forceinline__ void p_chain(const float* a_pre, const float* Cm, float* out,
                                        int nt, int lane) {
    int row = lane & 15, hi = lane >> 4;
    v8f acc = {0.f, 0.f, 0.f, 0.f, 0.f, 0.f, 0.f, 0.f};
#pragma unroll
    for (int kk = 0; kk < 16; ++kk) {
        int k0 = kk * 4 + hi * 2;
        v2f a, b;
        a.x = a_pre[2 * kk]; a.y = a_pre[2 * kk + 1];
        b.x = Cm[(k0    ) * 64 + nt * 16 + row];
        b.y = Cm[(k0 + 1) * 64 + nt * 16 + row];
        acc = __builtin_amdgcn_wmma_f32_16x16x4_f32(false, a, false, b, (short)0, acc, false, false);
    }
    if (lane < 16) {
#pragma unroll
        for (int r = 0; r < 8; ++r) out[r * 64 + nt * 16 + lane] = acc[r];
    }
}

// ---- constants: padded C, C^T, ord6 table --------------------------------
__global__ void k_init(const float* cooc, const float* Word, const float* bord, float* ws) {
    float* Cp = ws + OFF_CPAD;
    float* Ct = ws + OFF_CPADT;
    float* o6 = ws + OFF_ORD6;
    int tid = threadIdx.x;
    for (int e = tid; e < 4096; e += blockDim.x) {
        int k = e >> 6, n = e & 63;
        bool in = (k < DVEC) && (n < DVEC);
        Cp[e] = in ? cooc[k * DVEC + n] : 0.f;
        Ct[e] = in ? cooc[n * DVEC + k] : 0.f;
    }
    if (tid < 3) {
        float s = Word[tid] + bord[0];
        float l = 1.f / (1.f + expf(-s));
        o6[tid] = l; o6[tid + 3] = 1.f - l;
    }
}

// ---- leaves: softmax(emb[x]) * learn[x] + fixed[x]; init beam 0 ----------
__global__ void k_leaves(const int* x, const float* emb, const float* learn,
                         const float* fixedv, float* ws) {
    __shared__ float red[64];
    __shared__ float s_max, s_sum;
    int i = blockIdx.x, t = threadIdx.x;
    float* Vv = ws + OFF_VV;
    float* Vop = ws + OFF_OP;
    float* Vord = ws + OFF_ORD;
    int xi = x[i];
    float e = (t < DVEC) ? emb[(size_t)xi * DVEC + t] : -3.4e38f;
    red[t] = e; __syncthreads();
    if (t == 0) { float m = red[0]; for (int j = 1; j < DVEC; ++j) m = fmaxf(m, red[j]); s_max = m; }
    __syncthreads();
    float ex = (t < DVEC) ? expf(e - s_max) : 0.f;
    red[t] = ex; __syncthreads();
    if (t == 0) { float s = 0.f; for (int j = 0; j < DVEC; ++j) s += red[j]; s_sum = s; }
    __syncthreads();
    float xv = (t < DVEC) ? (ex / s_sum) * learn[xi] + fixedv[(size_t)xi * DVEC + t] : 0.f;
    Vv[((size_t)i * 8 + 0) * 64 + t] = xv;            // beam 0 (flags at 62/63 = 0)
#pragma unroll
    for (int b = 1; b < 8; ++b) Vv[((size_t)i * 8 + b) * 64 + t] = 0.f;
    if (t < 8) { Vop[i * 8 + t] = (t == 0) ? 1.f : TINYF; Vord[i * 8 + t] = (t == 0) ? 1.f : 0.f; }
}

// ---- P products for leaf row (entries ent(0,i)=i) ------------------------
__global__ void k_leafP(float* ws) {
    int i = blockIdx.x, lane = threadIdx.x;
    const float* Cp = ws + OFF_CPAD;
    const float* Ct = ws + OFF_CPADT;
    const float* vec = ws + OFF_VV + (size_t)i * 512;
    float* pf = ws + OFF_PF + (size_t)i * 512;
    float* pr = ws + OFF_PR + (size_t)i * 512;
    float a_pre[32];
    p_preload(vec, lane, a_pre);        // A fragments shared by all 8 chains
#pragma unroll
    for (int nt = 0; nt < 4; ++nt) {
        p_chain(a_pre, Cp, pf, nt, lane);
        p_chain(a_pre, Ct, pr, nt, lane);
    }
}

// ---- one DP step d: grid = 48-d blocks (span start i), 4 wave32 ----------
__global__ void __launch_bounds__(128) k_step(int d, float* ws) {
    __shared__ float sA[4][1024];   // per-wave: P_fwd rows 0-7, P_rev rows 8-15
    __shared__ float sC[4][512];    // per-wave: c-entry vectors (8x64)
    __shared__ float sOp[4][32];    // ob[8], oc[8], rb[8], rc[8]
    __shared__ float sL[4][16];     // weighted logs lb[8], lc[8]
    __shared__ float sFG[4][128];   // F[8][8], G[8][8]
    __shared__ float ms[1024]; __shared__ int msi[1024];
    __shared__ float mcp[1024]; __shared__ float mrd[1024];
    __shared__ float t2s[256]; __shared__ int t2i[256];
    __shared__ float t2c[256]; __shared__ float t2r[256];
    __shared__ float newvec[512];
    __shared__ int selI[8]; __shared__ float selC[8]; __shared__ float selR[8];

    const float* Cp = ws + OFF_CPAD;
    const float* Ct = ws + OFF_CPADT;
    const float* o6g = ws + OFF_ORD6;
    float* Vv = ws + OFF_VV;
    float* Pf = ws + OFF_PF;
    float* Pr = ws + OFF_PR;
    float* Vop = ws + OFF_OP;
    float* Vord = ws + OFF_ORD;

    int i = blockIdx.x;
    int tid = threadIdx.x, w = tid >> 5, lane = tid & 31;
    int row = lane & 15, hi = lane >> 4;
    float invd = 1.f / (float)d;

    float o6[6];
#pragma unroll
    for (int q = 0; q < 6; ++q) o6[q] = o6g[q];

    Top8 top; top8_reset(top);

    for (int k = w; k < d; k += 4) {                    // split points, wave-strided
        int eb = k * 48 + i;                            // left child (len k+1 at i)
        int ec = (d - 1 - k) * 48 + (i + k + 1);        // right child
        const float* gPf = Pf + (size_t)eb * 512;
        const float* gPr = Pr + (size_t)eb * 512;
        const float* gCv = Vv + (size_t)ec * 512;
        for (int t = lane; t < 512; t += 32) {
            sA[w][t]       = gPf[t];
            sA[w][512 + t] = gPr[t];
            sC[w][t]       = gCv[t];
        }
        if (lane < 8) {
            sOp[w][lane]      = Vop[eb * 8 + lane];
            sOp[w][8 + lane]  = Vop[ec * 8 + lane];
            sOp[w][16 + lane] = Vord[eb * 8 + lane];
            sOp[w][24 + lane] = Vord[ec * 8 + lane];
        }
        wave_lds_fence();
        float wb = (k == 0) ? 1.f : (float)k;                       // bf weights
        float wc = (d - 1 - k == 0) ? 1.f : (float)(d - 1 - k);     // cf weights
        if (lane < 8)       sL[w][lane] = wb * logf(sOp[w][lane]);
        else if (lane < 16) sL[w][lane] = wc * logf(sOp[w][lane]);
        wave_lds_fence();

        // D(16x16) = [P_fwd;P_rev](16x64) @ cvec^T(64x16): F rows 0-7, G rows 8-15.
        // B cols 8-15 duplicate cols 0-7 (branch-free); those D cols are unread.
        v8f acc = {0.f, 0.f, 0.f, 0.f, 0.f, 0.f, 0.f, 0.f};
        const float* Arow = &sA[w][row * 64];
        const float* Bcol = &sC[w][(lane & 7) * 64];
#pragma unroll
        for (int kk = 0; kk < 16; ++kk) {
            int k0 = kk * 4 + hi * 2;
            v2f a, b;
            a.x = Arow[k0]; a.y = Arow[k0 + 1];
            b.x = Bcol[k0]; b.y = Bcol[k0 + 1];
            acc = __builtin_amdgcn_wmma_f32_16x16x4_f32(false, a, false, b, (short)0, acc, false, false);
        }
        if (lane < 8) {                                  // F[r][col=lane]
#pragma unroll
            for (int r = 0; r < 8; ++r) sFG[w][r * 8 + lane] = acc[r];
        } else if (lane >= 16 && lane < 24) {            // G[r][col=lane-16]
#pragma unroll
            for (int r = 0; r < 8; ++r) sFG[w][64 + r * 8 + (lane - 16)] = acc[r];
        }
        wave_lds_fence();

        // 384 candidates: ops [logF,logF,logG | logG,logG,logF], compose rules below
        for (int t = lane; t < 384; t += 32) {
            int pair = t / 6, op = t - pair * 6;
            int b = pair >> 3, c = pair & 7;
            float val = (op == 0 || op == 1 || op == 5) ? sFG[w][b * 8 + c]
                                                        : sFG[w][64 + b * 8 + c];
            float lsc  = logf(val + TINYF) + sL[w][b] + sL[w][8 + c];
            float cop  = expf(lsc * invd);
            float cord = sOp[w][16 + b] * sOp[w][24 + c] * o6[op];
            top8_ins(top, cop * cord, k * 384 + t, cop, cord);
        }
        wave_lds_fence();
    }

    // hierarchical exact top-8 merge: 128x8 -> 32x8 -> 8
    __syncthreads();
    {
        int base = tid * 8;
#pragma unroll
        for (int q = 0; q < 8; ++q) { ms[base+q]=top.s[q]; msi[base+q]=top.i[q]; mcp[base+q]=top.c[q]; mrd[base+q]=top.r[q]; }
    }
    __syncthreads();
    if (tid < 32) {
        top8_reset(top);
        for (int e = tid * 32; e < tid * 32 + 32; ++e) top8_ins(top, ms[e], msi[e], mcp[e], mrd[e]);
        int base = tid * 8;
#pragma unroll
        for (int q = 0; q < 8; ++q) { t2s[base+q]=top.s[q]; t2i[base+q]=top.i[q]; t2c[base+q]=top.c[q]; t2r[base+q]=top.r[q]; }
    }
    __syncthreads();
    if (tid == 0) {
        top8_reset(top);
        for (int e = 0; e < 256; ++e) top8_ins(top, t2s[e], t2i[e], t2c[e], t2r[e]);
#pragma unroll
        for (int q = 0; q < 8; ++q) { selI[q] = top.i[q]; selC[q] = top.c[q]; selR[q] = top.r[q]; }
    }
    __syncthreads();

    // reconstruct selected vectors (compose op rules) and commit V[d][i]
    int entD = d * 48 + i;
    for (int e = tid; e < 512; e += 128) {
        int s = e >> 6, j = e & 63;
        int idx = selI[s];
        int k = idx / 384, rem = idx - k * 384;
        int pair = rem / 6, op = rem - pair * 6;
        int b = pair >> 3, c = pair & 7;
        const float* bv = Vv + ((size_t)(k * 48 + i) * 8 + b) * 64;
        const float* cv = Vv + ((size_t)((d - 1 - k) * 48 + i + k + 1) * 8 + c) * 64;
        float v;
        if      (op == 0) v = (j == 62) ? 1.f : ((j == 63) ? ((bv[63] != 0.f) ? 1.f : 0.f) : bv[j]);
        else if (op == 1) v = (j == 63) ? 1.f : ((j == 62) ? ((bv[62] != 0.f) ? 1.f : 0.f) : bv[j]);
        else if (op == 2) v = cv[j];
        else if (op == 3) v = (j == 62) ? 1.f : ((j == 63) ? ((cv[63] != 0.f) ? 1.f : 0.f) : cv[j]);
        else if (op == 4) v = (j == 63) ? 1.f : ((j == 62) ? ((cv[62] != 0.f) ? 1.f : 0.f) : cv[j]);
        else              v = bv[j];
        newvec[e] = v;
        Vv[(size_t)entD * 512 + e] = v;
    }
    if (tid < 8) { Vop[entD * 8 + tid] = selC[tid]; Vord[entD * 8 + tid] = selR[tid]; }
    __syncthreads();

    // P products for the new entry: wave w owns N-tile w (both directions)
    {
        float a_pre[32];
        p_preload((const float*)newvec, lane, a_pre);
        p_chain(a_pre, Cp, Pf + (size_t)entD * 512, w, lane);
        p_chain(a_pre, Ct, Pr + (size_t)entD * 512, w, lane);
    }
}

__global__ void k_out(const float* ws, float* out) {
    int t = threadIdx.x;
    const float* Vop = ws + OFF_OP;
    const float* Vord = ws + OFF_ORD;
    if (t < 8) out[t] = Vop[(47 * 48) * 8 + t] * Vord[(47 * 48) * 8 + t];
}

extern "C" void kernel_launch(void* const* d_in, const int* in_sizes, int n_in,
                              void* d_out, int out_size, void* d_ws, size_t ws_size,
                              hipStream_t stream) {
    const int*   x      = (const int*)d_in[0];
    const float* emb    = (const float*)d_in[1];
    const float* learn  = (const float*)d_in[2];
    const float* fixedv = (const float*)d_in[3];
    const float* cooc   = (const float*)d_in[4];
    const float* Word   = (const float*)d_in[5];
    const float* bord   = (const float*)d_in[6];
    float* ws = (float*)d_ws;   // needs ~14.4 MB

    k_init  <<<1, 256, 0, stream>>>(cooc, Word, bord, ws);
    k_leaves<<<48, 64, 0, stream>>>(x, emb, learn, fixedv, ws);
    k_leafP <<<48, 32, 0, stream>>>(ws);
    for (int d = 1; d < 48; ++d)
        k_step<<<48 - d, 128, 0, stream>>>(d, ws);
    k_out   <<<1, 32, 0, stream>>>(ws, (float*)d_out);
}